// RNN1_28733331210586
// MI455X (gfx1250) — compile-verified
//
#include <hip/hip_runtime.h>
#include <cmath>

typedef float v2f __attribute__((ext_vector_type(2)));
typedef float v8f __attribute__((ext_vector_type(8)));

#define B_ 4096
#define T_ 256
#define E_ 27
#define H_ 64
#define WE_ (H_ + E_)   /* 91: row stride of Wa */
#define LDSTRIDE 68     /* 64 + 4 pad -> bank-conflict-free A-frag loads */

// Branch-free tanh: CDNA5 hardware v_tanh_f32 (confirmed emitted), else exp2+rcp.
__device__ __forceinline__ float fast_tanh(float x) {
#if __has_builtin(__builtin_amdgcn_tanhf)
    return __builtin_amdgcn_tanhf(x);
#elif __has_builtin(__builtin_amdgcn_tanh_f32)
    return __builtin_amdgcn_tanh_f32(x);
#else
    float e = __builtin_amdgcn_exp2f(x * 2.885390081777927f);
    float r = __builtin_amdgcn_rcpf(e + 1.0f);
    return 1.0f - 2.0f * r;
#endif
}

__global__ __launch_bounds__(128)
void rnn1_fwd_wmma(const float* __restrict__ word, const float* __restrict__ Wa,
                   const float* __restrict__ ba,   const float* __restrict__ Wy,
                   const float* __restrict__ by,   float* __restrict__ out)
{
    __shared__ float a_lds[16 * LDSTRIDE];

    const int tid  = threadIdx.x;
    const int lane = tid & 31;
    const int wv   = tid >> 5;          // wave id 0..3 = hidden N-tile
    const int hl   = lane & 15;         // 0..15
    const int kh   = (lane >> 4) << 1;  // 0 or 2 : K sub-offset (A/B frag layout)
    const int b0   = blockIdx.x * 16;   // batch tile origin

    // ---------------- weight B-fragments, resident in registers ----------------
    // B[k][n] layout mirrors A: vec elem j holds K = 4*kstep + j + kh, N = tile*16 + hl
    v2f WaaF[16], WaxF[7], WyF[16];
    const int na = wv * 16 + hl;            // hidden column owned by this wave
    #pragma unroll
    for (int k = 0; k < 16; ++k) {
        int k0 = 4 * k + kh;
        WaaF[k][0] = Wa[na * WE_ + k0 + 0]; // Waa = Wa[:, :64]; B = Waa^T
        WaaF[k][1] = Wa[na * WE_ + k0 + 1];
    }
    #pragma unroll
    for (int k = 0; k < 7; ++k) {           // K padded 27 -> 28; pad col = 0
        int k0 = 4 * k + kh;
        WaxF[k][0] = (k0 + 0 < E_) ? Wa[na * WE_ + H_ + k0 + 0] : 0.f;
        WaxF[k][1] = (k0 + 1 < E_) ? Wa[na * WE_ + H_ + k0 + 1] : 0.f;
    }
    const int ny = (wv & 1) * 16 + hl;      // output column for y N-tiles (waves 0,1)
    #pragma unroll
    for (int k = 0; k < 16; ++k) {          // N padded 27 -> 32; pad cols = 0
        int k0 = 4 * k + kh;
        WyF[k][0] = (ny < E_) ? Wy[ny * H_ + k0 + 0] : 0.f;
        WyF[k][1] = (ny < E_) ? Wy[ny * H_ + k0 + 1] : 0.f;
    }
    const float bav = ba[na];
    const float byv = (ny < E_) ? by[ny] : 0.f;

    // a_0 = 0
    for (int i = tid; i < 16 * LDSTRIDE; i += 128) a_lds[i] = 0.f;

    // A-fragments of a_t (invariant at loop top); a_0 = 0
    v2f aF[16];
    #pragma unroll
    for (int k = 0; k < 16; ++k) { aF[k][0] = 0.f; aF[k][1] = 0.f; }

    // per-lane row base into word (A-frag rows map to contiguous float2 pairs)
    const float* wrow = word + (size_t)(b0 + hl) * (T_ * E_) + kh;
    const int wvu = __builtin_amdgcn_readfirstlane(wv); // scalar branch (WMMA needs EXEC all-1)

    __syncthreads();

    for (int t = 0; t < T_; ++t) {
        // ---- x = word[:, t-1, :] A-fragments straight from global ----
        v2f xF[7];
        if (t > 0) {
            const float* xp = wrow + (size_t)(t - 1) * E_;
            #pragma unroll
            for (int k = 0; k < 7; ++k) {   // k=6 reads elem 27 (next row) * WaxF==0 -> harmless
                xF[k][0] = xp[4 * k + 0];
                xF[k][1] = xp[4 * k + 1];
            }
        }
        if (t + 1 < T_) __builtin_prefetch(wrow + (size_t)t * E_, 0, 3);

        // ---- a' = tanh(a @ Waa^T + x @ Wax^T + ba), this wave's 16 columns ----
        // Two independent accumulator chains (even/odd K) to break the D->C
        // dependency chain; merged with 8 fp32 adds at the end.
        v8f acc0, acc1;
        #pragma unroll
        for (int v = 0; v < 8; ++v) { acc0[v] = bav; acc1[v] = 0.f; }
        #pragma unroll
        for (int k = 0; k < 8; ++k) {
            acc0 = __builtin_amdgcn_wmma_f32_16x16x4_f32(
                       false, aF[2*k+0], false, WaaF[2*k+0], (short)0, acc0, false, false);
            acc1 = __builtin_amdgcn_wmma_f32_16x16x4_f32(
                       false, aF[2*k+1], false, WaaF[2*k+1], (short)0, acc1, false, false);
        }
        if (t > 0) {
            #pragma unroll
            for (int k = 0; k < 6; k += 2) {
                acc0 = __builtin_amdgcn_wmma_f32_16x16x4_f32(
                           false, xF[k+0], false, WaxF[k+0], (short)0, acc0, false, false);
                acc1 = __builtin_amdgcn_wmma_f32_16x16x4_f32(
                           false, xF[k+1], false, WaxF[k+1], (short)0, acc1, false, false);
            }
            acc0 = __builtin_amdgcn_wmma_f32_16x16x4_f32(
                       false, xF[6], false, WaxF[6], (short)0, acc0, false, false);
        }
        v8f acc;
        #pragma unroll
        for (int v = 0; v < 8; ++v) acc[v] = fast_tanh(acc0[v] + acc1[v]);

        __syncthreads();                    // everyone done with a_t fragments
        // D-layout -> LDS: VGPR v holds (row = v + 8*(lane/16), col = na)
        #pragma unroll
        for (int v = 0; v < 8; ++v) {
            int row = v + ((lane >> 4) << 3);
            a_lds[row * LDSTRIDE + na] = acc[v];
        }
        __syncthreads();                    // a_{t+1} tile visible to all waves
        // reload A-fragments of a_{t+1} (used by y now and by next step's recurrence)
        #pragma unroll
        for (int k = 0; k < 16; ++k) {
            const float* p = &a_lds[hl * LDSTRIDE + 4 * k + kh];
            aF[k][0] = p[0];
            aF[k][1] = p[1];
        }

        // ---- y_t = a_{t+1} @ Wy^T + by : waves 0,1 each own one 16-col N-tile ----
        if (wvu < 2) {
            v8f accy0, accy1;
            #pragma unroll
            for (int v = 0; v < 8; ++v) { accy0[v] = byv; accy1[v] = 0.f; }
            #pragma unroll
            for (int k = 0; k < 8; ++k) {
                accy0 = __builtin_amdgcn_wmma_f32_16x16x4_f32(
                            false, aF[2*k+0], false, WyF[2*k+0], (short)0, accy0, false, false);
                accy1 = __builtin_amdgcn_wmma_f32_16x16x4_f32(
                            false, aF[2*k+1], false, WyF[2*k+1], (short)0, accy1, false, false);
            }
            if (ny < E_) {                  // mask pad columns 27..31
                #pragma unroll
                for (int v = 0; v < 8; ++v) {
                    int row = v + ((lane >> 4) << 3);
                    out[(size_t)(b0 + row) * (T_ * E_) + (size_t)t * E_ + ny] =
                        accy0[v] + accy1[v];
                }
            }
        }
    }
}

extern "C" void kernel_launch(void* const* d_in, const int* in_sizes, int n_in,
                              void* d_out, int out_size, void* d_ws, size_t ws_size,
                              hipStream_t stream) {
    const float* word = (const float*)d_in[0];
    const float* Wa   = (const float*)d_in[1];
    const float* ba   = (const float*)d_in[2];
    const float* Wy   = (const float*)d_in[3];
    const float* by   = (const float*)d_in[4];
    float* out = (float*)d_out;
    rnn1_fwd_wmma<<<dim3(B_ / 16), dim3(128), 0, stream>>>(word, Wa, ba, Wy, by, out);
}